// QNN_37726992728716
// MI455X (gfx1250) — compile-verified
//
#include <hip/hip_runtime.h>
#include <stdint.h>

#define NQ      21
#define TSH     13
#define TS      (1 << TSH)     // 8192 amplitudes per tile (64 KB LDS as float2)
#define THREADS 512
#define NBLK    256            // 2^21 / 2^13

typedef unsigned int u32x4 __attribute__((ext_vector_type(4)));
typedef int          i32x8 __attribute__((ext_vector_type(8)));
typedef int          i32x4 __attribute__((ext_vector_type(4)));

#if defined(__has_builtin)
#if __has_builtin(__builtin_amdgcn_tensor_load_to_lds) && __has_builtin(__builtin_amdgcn_s_wait_tensorcnt)
#define USE_TDM_LOAD 1
#endif
#if __has_builtin(__builtin_amdgcn_tensor_store_from_lds) && __has_builtin(__builtin_amdgcn_s_wait_tensorcnt)
#define USE_TDM_STORE 1
#endif
#endif
#ifndef USE_TDM_LOAD
#define USE_TDM_LOAD 0
#endif
#ifndef USE_TDM_STORE
#define USE_TDM_STORE 0
#endif

#if USE_TDM_LOAD || USE_TDM_STORE
// Build a TDM D# per CDNA5 ISA.
// 2D tile: tile_d0 contiguous 8-byte elements per row, tile_d1 rows, row stride
// stride0 (elements). Tile is contiguous in LDS starting at lds_byte.
// This toolchain's builtins take 6 args: D# groups 0..3 (4+8+4+4 dwords), an
// extra 8-dword group (beyond the documented encoding; VADDR4 is unused by the ISA
// -> zero-filled), and the cache-policy immediate.
struct TdmDesc { u32x4 g0; i32x8 g1; i32x4 g2; i32x4 g3; i32x8 g4; };

__device__ __forceinline__ TdmDesc tdm_desc(uint32_t lds_byte, const void* gptr,
                                            uint32_t tile_d0, uint32_t tile_d1,
                                            uint32_t stride0) {
  TdmDesc d;
  uint64_t ga = (uint64_t)(uintptr_t)gptr;
  // Group 0: count=1 | lds_addr | global_addr[56:0] | type=2 (bits 127:126)
  d.g0 = u32x4{ 1u, lds_byte, (uint32_t)ga,
                (uint32_t)((ga >> 32) & 0x1FFFFFFu) | 0x80000000u };
  const uint32_t td0 = 1u << 21, td1 = 1u << 21;   // tensor dims large: no OOB
  // Group 1: data_size=3 (8B); tensor_dim0 @bits79:48; tensor_dim1 @111:80;
  // tile_dim0 @127:112; tile_dim1 @143:128; tile_dim2=1 @159:144;
  // tensor_dim0_stride @207:160 (48b); tensor_dim1_stride=0 (tile_dim2==1).
  d.g1 = i32x8{ (int)(3u << 16),
                (int)((td0 & 0xFFFFu) << 16),
                (int)((td0 >> 16) | ((td1 & 0xFFFFu) << 16)),
                (int)((td1 >> 16) | (tile_d0 << 16)),
                (int)(tile_d1 | (1u << 16)),
                (int)stride0, 0, 0 };
  // Group 2: tensor_dim2=1, tensor_dim3=1, stride2=0, tile_dim3=1
  d.g2 = i32x4{ 1, 1, 0, (int)(1u << 16) };
  // Group 3: stride3=0, tensor_dim4=1 (@79:48), tile_dim4=1 (@95:80)
  d.g3 = i32x4{ 0, (int)(1u << 16), (int)(1u << 16), 0 };
  d.g4 = i32x8{ 0, 0, 0, 0, 0, 0, 0, 0 };
  return d;
}
#endif

#if USE_TDM_LOAD
__device__ __forceinline__ void tdm_load_tile(uint32_t lds_byte, const void* gptr,
                                              uint32_t tile_d0, uint32_t tile_d1,
                                              uint32_t stride0) {
  TdmDesc d = tdm_desc(lds_byte, gptr, tile_d0, tile_d1, stride0);
  __builtin_amdgcn_tensor_load_to_lds(d.g0, d.g1, d.g2, d.g3, d.g4, 0);
}
#endif
#if USE_TDM_STORE
__device__ __forceinline__ void tdm_store_tile(uint32_t lds_byte, void* gptr,
                                               uint32_t tile_d0, uint32_t tile_d1,
                                               uint32_t stride0) {
  TdmDesc d = tdm_desc(lds_byte, gptr, tile_d0, tile_d1, stride0);
  __builtin_amdgcn_tensor_store_from_lds(d.g0, d.g1, d.g2, d.g3, d.g4, 0);
}
#endif

__device__ __forceinline__ bool is_wave0(int tid) {
  return (__builtin_amdgcn_readfirstlane(tid) >> 5) == 0;  // scalar branch
}

// Rot(phi,theta,omega) = RZ(omega) RY(theta) RZ(phi):
// U00=ep*c  U01=-em*s  U10=conj(em)*s  U11=conj(ep)*c
// ep=exp(-0.5i(phi+omega)), em=exp(+0.5i(phi-omega)), c=cos(th/2), s=sin(th/2)
__device__ __forceinline__ void rot_entries(const float* __restrict__ p, float2* U) {
  float phi = p[0], th = p[1], om = p[2];
  float c = cosf(0.5f * th), s = sinf(0.5f * th);
  float a = -0.5f * (phi + om), b = 0.5f * (phi - om);
  float ca = cosf(a), sa = sinf(a), cb = cosf(b), sb = sinf(b);
  U[0] = make_float2(ca * c,  sa * c);
  U[1] = make_float2(-cb * s, -sb * s);
  U[2] = make_float2(cb * s,  -sb * s);
  U[3] = make_float2(ca * c,  -sa * c);
}

// Apply 2x2 gate across LDS-index bit b (pairs i0, i0|1<<b).
__device__ __forceinline__ void lds_rot(float2* tile, int b, float2 U0, float2 U1,
                                        float2 U2, float2 U3, int tid) {
  const int bm = (1 << b) - 1;
  for (int t = tid; t < TS / 2; t += THREADS) {
    int i0 = ((t & ~bm) << 1) | (t & bm);
    int i1 = i0 | (1 << b);
    float2 a0 = tile[i0], a1 = tile[i1];
    float2 n0, n1;
    n0.x = U0.x * a0.x - U0.y * a0.y + U1.x * a1.x - U1.y * a1.y;
    n0.y = U0.x * a0.y + U0.y * a0.x + U1.x * a1.y + U1.y * a1.x;
    n1.x = U2.x * a0.x - U2.y * a0.y + U3.x * a1.x - U3.y * a1.y;
    n1.y = U2.x * a0.y + U2.y * a0.x + U3.x * a1.y + U3.y * a1.x;
    tile[i0] = n0;
    tile[i1] = n1;
  }
  __syncthreads();
}

// One whole CNOT sub-layer (all controls = target_bit+1) as a single involutive
// permutation: j = i ^ ((i>>1) & targetMask); swap once per unordered pair.
__device__ __forceinline__ void lds_cnot_layer(float2* tile, int mask, int tid) {
  for (int i = tid; i < TS; i += THREADS) {
    int j = i ^ ((i >> 1) & mask);
    if (j > i) { float2 tmp = tile[i]; tile[i] = tile[j]; tile[j] = tmp; }
  }
  __syncthreads();
}

// ---- Pass A: contiguous tiles, flat bits 0..12 local (qubits 8..20) ----
// Gates: Rot(q=8..20) [bit 12-g], even CNOT targets {11,9,7,5,3,1}=0xAAA,
// odd CNOT targets {10,8,6,4,2,0}=0x555.
__global__ void __launch_bounds__(THREADS)
qsim_passA(const float* __restrict__ feat, float2* __restrict__ state,
           const float* __restrict__ param, int layer, int from_feat) {
  extern __shared__ float2 tile[];
  __shared__ float2 G[13][4];
  const int tid = threadIdx.x;
  const int base = blockIdx.x * TS;

  if (tid < 13) rot_entries(&param[(layer * NQ + (8 + tid)) * 3], G[tid]);

  if (from_feat) {
    const float4* f4 = (const float4*)(feat + base);
    for (int t = tid; t < TS / 4; t += THREADS) {
      float4 v = f4[t];
      tile[t * 4 + 0] = make_float2(v.x, 0.f);
      tile[t * 4 + 1] = make_float2(v.y, 0.f);
      tile[t * 4 + 2] = make_float2(v.z, 0.f);
      tile[t * 4 + 3] = make_float2(v.w, 0.f);
    }
  } else {
#if USE_TDM_LOAD
    if (is_wave0(tid)) {
      tdm_load_tile((uint32_t)(uintptr_t)tile, state + base, TS, 1, TS);
      __builtin_amdgcn_s_wait_tensorcnt(0);
    }
#else
    const float4* s4 = (const float4*)(state + base);
    float4* t4 = (float4*)tile;
    for (int t = tid; t < TS / 2; t += THREADS) t4[t] = s4[t];
#endif
  }
  __syncthreads();

  for (int g = 0; g < 13; ++g)
    lds_rot(tile, 12 - g, G[g][0], G[g][1], G[g][2], G[g][3], tid);
  lds_cnot_layer(tile, 0x0AAA, tid);  // even: (8,9)(10,11)(12,13)(14,15)(16,17)(18,19)
  lds_cnot_layer(tile, 0x0555, tid);  // odd:  (9,10)(11,12)(13,14)(15,16)(17,18)(19,20)

#if USE_TDM_STORE
  // All LDS writes DS-complete at the barrier above; wave 0 hands the tile to
  // the TDM. S_ENDPGM's implicit wait-idle covers TENSORcnt drain.
  if (is_wave0(tid))
    tdm_store_tile((uint32_t)(uintptr_t)tile, state + base, TS, 1, TS);
#else
  float4* s4 = (float4*)(state + base);
  const float4* t4 = (const float4*)tile;
  for (int t = tid; t < TS / 2; t += THREADS) s4[t] = t4[t];
#endif
}

// ---- Pass B: strided tiles. rows = bits 12..20 (512), cols = bits 0..3 (16),
// block fixes bits 4..11. LDS index = row*16+col; global bit k -> LDS bit k-8.
// Gates: Rot(q=0..7) [LDS bit 12-q], even CNOT targets {11,9,7,5}=0xAA0,
// odd CNOT targets {10,8,6,4}=0x550.
__global__ void __launch_bounds__(THREADS)
qsim_passB(float2* __restrict__ state, const float* __restrict__ param,
           int layer, int do_reduce, float* __restrict__ partials) {
  extern __shared__ float2 tile[];
  __shared__ float2 G[8][4];
  __shared__ float red[THREADS];
  const int tid = threadIdx.x;
  float2* gbase = state + blockIdx.x * 16;

  if (tid < 8) rot_entries(&param[(layer * NQ + tid) * 3], G[tid]);

#if USE_TDM_LOAD
  if (is_wave0(tid)) {
    tdm_load_tile((uint32_t)(uintptr_t)tile, gbase, 16, 512, 4096);
    __builtin_amdgcn_s_wait_tensorcnt(0);
  }
#else
  for (int t = tid; t < TS; t += THREADS)
    tile[t] = gbase[(t >> 4) * 4096 + (t & 15)];
#endif
  __syncthreads();

  for (int g = 0; g < 8; ++g)
    lds_rot(tile, 12 - g, G[g][0], G[g][1], G[g][2], G[g][3], tid);
  lds_cnot_layer(tile, 0x0AA0, tid);  // even: (0,1)(2,3)(4,5)(6,7)
  lds_cnot_layer(tile, 0x0550, tid);  // odd:  (1,2)(3,4)(5,6)(7,8)

  if (!do_reduce) {
#if USE_TDM_STORE
    if (is_wave0(tid))
      tdm_store_tile((uint32_t)(uintptr_t)tile, gbase, 16, 512, 4096);
#else
    for (int t = tid; t < TS; t += THREADS)
      gbase[(t >> 4) * 4096 + (t & 15)] = tile[t];
#endif
  } else {
    // amp0: qubit 20 (= flat bit 0 = col bit 0) in |0> -> even t.
    float acc = 0.f;
    for (int t = tid; t < TS; t += THREADS) {
      if (!(t & 1)) { float2 v = tile[t]; acc += v.x * v.x + v.y * v.y; }
    }
    red[tid] = acc;
    __syncthreads();
    for (int s = THREADS / 2; s > 0; s >>= 1) {
      if (tid < s) red[tid] += red[tid + s];
      __syncthreads();
    }
    if (tid == 0) partials[blockIdx.x] = red[0];
  }
}

__global__ void qsim_reduce(const float* __restrict__ partials, float* __restrict__ out) {
  if (threadIdx.x == 0 && blockIdx.x == 0) {
    float s = 0.f;
    for (int i = 0; i < NBLK; ++i) s += partials[i];
    out[0] = s;
  }
}

extern "C" void kernel_launch(void* const* d_in, const int* in_sizes, int n_in,
                              void* d_out, int out_size, void* d_ws, size_t ws_size,
                              hipStream_t stream) {
  (void)in_sizes; (void)n_in; (void)out_size; (void)ws_size;
  const float* feat  = (const float*)d_in[0];   // 2^21 f32
  const float* param = (const float*)d_in[1];   // 3*21*3 f32
  float* out = (float*)d_out;
  float2* state = (float2*)d_ws;                // 2^21 complex64 = 16 MB
  float* partials = (float*)((char*)d_ws + ((size_t)1 << 21) * sizeof(float2));

  const size_t lds = TS * sizeof(float2);       // 64 KB dynamic LDS
  qsim_passA<<<NBLK, THREADS, lds, stream>>>(feat, state, param, 0, 1);
  qsim_passB<<<NBLK, THREADS, lds, stream>>>(state, param, 0, 0, nullptr);
  qsim_passA<<<NBLK, THREADS, lds, stream>>>(nullptr, state, param, 1, 0);
  qsim_passB<<<NBLK, THREADS, lds, stream>>>(state, param, 1, 0, nullptr);
  qsim_passA<<<NBLK, THREADS, lds, stream>>>(nullptr, state, param, 2, 0);
  qsim_passB<<<NBLK, THREADS, lds, stream>>>(state, param, 2, 1, partials);
  qsim_reduce<<<1, 64, 0, stream>>>(partials, out);
}